// fPEPS_Model_GPU_34153579938263
// MI455X (gfx1250) — compile-verified
//
#include <hip/hip_runtime.h>

// ---------------------------------------------------------------------------
// fPEPS amplitude: boundary-MPS contraction with SVD compression (chi=32),
// fp32 end-to-end. All GEMM-shaped work uses V_WMMA_F32_16X16X4_F32.
// One workgroup (8 wave32) per sample.
// ---------------------------------------------------------------------------

#define LXDIM 8
#define LYDIM 8
#define DDIM  8
#define PDIM  2
#define CHI   32
#define NT    256
#define NW    8

// per-sample workspace layout (floats)
#define SLOT      (256 * 256 * 8)        // max MPS tensor (L,R,8)
#define ARENA     (8 * SLOT)
#define OFF_BOT   0
#define OFF_TOP   (ARENA)
#define OFF_M     (2 * ARENA)
#define OFF_U     (2 * ARENA + SLOT)
#define OFF_G     (2 * ARENA + 2 * SLOT)
#define OFF_EV    (OFF_G + 65536)
#define OFF_W     (OFF_EV + 65536)
#define OFF_SB    (OFF_W + 65536)        // 8192 floats
#define OFF_SITE  (OFF_SB + 8192)        // 8 * 4096
#define SAMPLE_F  (OFF_SITE + 32768)

typedef __attribute__((ext_vector_type(2))) float v2f;
typedef __attribute__((ext_vector_type(8))) float v8f;

// ---------------------------------------------------------------------------
// Block-cooperative GEMM, stride-generic:
//   C(i,j) = sum_k A(i,k)*B(k,j)
//   A(i,k)=A[i*sA0+k*sA1], B(k,j)=B[k*sB0+j*sB1], C(i,j)=C[i*sC0+j*sC1]
// Each wave computes whole 16x16 tiles with V_WMMA_F32_16X16X4_F32.
// Full interior tiles take an unguarded fast path (no branches in the K loop);
// edge tiles use clamped addresses + value selects (branch-free loads).
// ---------------------------------------------------------------------------
__device__ void block_gemm(float* __restrict__ C,
                           const float* __restrict__ A,
                           const float* __restrict__ B,
                           int M, int N, int K,
                           int sA0, int sA1, int sB0, int sB1,
                           int sC0, int sC1)
{
    const int lane = threadIdx.x & 31;
    const int wave = threadIdx.x >> 5;
    const int half = lane >> 4;     // 0: lanes 0-15, 1: lanes 16-31
    const int lr   = lane & 15;
    const int tm = (M + 15) >> 4;
    const int tn = (N + 15) >> 4;
    const int Kc = K - 1;
    for (int t = wave; t < tm * tn; t += NW) {
        const int ti = t / tn;
        const int tj = t - ti * tn;
        const int row = ti * 16 + lr;   // A row for this lane
        const int col = tj * 16 + lr;   // B col for this lane
        v8f acc = {0.f, 0.f, 0.f, 0.f, 0.f, 0.f, 0.f, 0.f};
        // wave-uniform fast-path test (identical for all lanes of the wave)
        const bool full = (ti * 16 + 16 <= M) && (tj * 16 + 16 <= N) &&
                          ((K & 3) == 0);
        if (full) {
            const float* Ar = A + row * sA0;
            const float* Bc = B + col * sB1;
            for (int k0 = 0; k0 < K; k0 += 4) {
                const int ka = k0 + half * 2;
                v2f af, bf;
                af.x = Ar[ka * sA1];
                af.y = Ar[(ka + 1) * sA1];
                bf.x = Bc[ka * sB0];
                bf.y = Bc[(ka + 1) * sB0];
                __builtin_prefetch(Ar + (ka + 4) * sA1, 0, 1);
                __builtin_prefetch(Bc + (ka + 4) * sB0, 0, 1);
                acc = __builtin_amdgcn_wmma_f32_16x16x4_f32(
                    false, af, false, bf, (short)0, acc, false, false);
            }
            float* Cc = C + (ti * 16 + half * 8) * sC0 + col * sC1;
#pragma unroll
            for (int v = 0; v < 8; ++v) Cc[v * sC0] = acc[v];
        } else {
            const bool rOK = row < M;
            const bool cOK = col < N;
            const float* Ar = A + (rOK ? row : M - 1) * sA0;
            const float* Bc = B + (cOK ? col : N - 1) * sB1;
            for (int k0 = 0; k0 < K; k0 += 4) {
                const int ka = k0 + half * 2;
                const int kb = ka + 1;
                const int kac = ka < K ? ka : Kc;
                const int kbc = kb < K ? kb : Kc;
                // unconditional (clamped) loads, branch-free zero masking
                const float a0 = Ar[kac * sA1];
                const float a1 = Ar[kbc * sA1];
                const float b0 = Bc[kac * sB0];
                const float b1 = Bc[kbc * sB0];
                v2f af, bf;
                af.x = (rOK && ka < K) ? a0 : 0.f;
                af.y = (rOK && kb < K) ? a1 : 0.f;
                bf.x = (cOK && ka < K) ? b0 : 0.f;
                bf.y = (cOK && kb < K) ? b1 : 0.f;
                acc = __builtin_amdgcn_wmma_f32_16x16x4_f32(
                    false, af, false, bf, (short)0, acc, false, false);
            }
            const int mbase = ti * 16 + half * 8;
            if (cOK) {
#pragma unroll
                for (int v = 0; v < 8; ++v) {
                    const int mr = mbase + v;
                    if (mr < M) C[mr * sC0 + col * sC1] = acc[v];
                }
            }
        }
    }
}

__device__ void block_copy(float* dst, const float* src, int count)
{
    for (int i = threadIdx.x; i < count; i += NT) dst[i] = src[i];
}

// ---------------------------------------------------------------------------
// Symmetric Jacobi eigensolver, G (n x n) -> approx diag(G), V columns = evecs.
// Tournament (round-robin) disjoint pair rotations, block-parallel.
// ---------------------------------------------------------------------------
__device__ void block_jacobi(float* G, float* V, int n,
                             float* shc, float* shs, int* shp, int* shq)
{
    const int tid = threadIdx.x;
    for (int i = tid; i < n * n; i += NT) V[i] = 0.f;
    __syncthreads();
    for (int i = tid; i < n; i += NT) V[i * n + i] = 1.f;
    __syncthreads();
    if (n == 1) return;
    const int np = n >> 1;
    for (int sweep = 0; sweep < 10; ++sweep) {
        for (int r = 0; r < n - 1; ++r) {
            if (tid < np) {
                int a = (tid == 0) ? 0 : (1 + ((tid - 1 + r) % (n - 1)));
                int bslot = n - 1 - tid;
                int b = 1 + ((bslot - 1 + r) % (n - 1));
                int p = a < b ? a : b;
                int q = a < b ? b : a;
                shp[tid] = p; shq[tid] = q;
                float apq = G[p * n + q];
                float c = 1.f, s = 0.f;
                if (fabsf(apq) > 1e-30f) {
                    float app = G[p * n + p];
                    float aqq = G[q * n + q];
                    float tau = (aqq - app) / (2.f * apq);
                    float tt = (tau >= 0.f ? 1.f : -1.f) /
                               (fabsf(tau) + sqrtf(1.f + tau * tau));
                    c = 1.f / sqrtf(1.f + tt * tt);
                    s = tt * c;
                }
                shc[tid] = c; shs[tid] = s;
            }
            __syncthreads();
            // G <- J^T G   (rows p,q)
            for (int id = tid; id < np * n; id += NT) {
                int k = id / n, j = id - k * n;
                int p = shp[k], q = shq[k];
                float c = shc[k], s = shs[k];
                float gp = G[p * n + j], gq = G[q * n + j];
                G[p * n + j] = c * gp - s * gq;
                G[q * n + j] = s * gp + c * gq;
            }
            __syncthreads();
            // G <- G J ; V <- V J   (cols p,q)
            for (int id = tid; id < np * n; id += NT) {
                int k = id / n, j = id - k * n;
                int p = shp[k], q = shq[k];
                float c = shc[k], s = shs[k];
                float gp = G[j * n + p], gq = G[j * n + q];
                G[j * n + p] = c * gp - s * gq;
                G[j * n + q] = s * gp + c * gq;
                float vp = V[j * n + p], vq = V[j * n + q];
                V[j * n + p] = c * vp - s * vq;
                V[j * n + q] = s * vp + c * vq;
            }
            __syncthreads();
        }
    }
}

// even-odd transposition sort of eigenvalues (descending), ord = permutation
__device__ void sort_eigs(const float* G, int n, float* lam, int* ord)
{
    const int tid = threadIdx.x;
    for (int i = tid; i < n; i += NT) { lam[i] = G[i * n + i]; ord[i] = i; }
    __syncthreads();
    for (int ph = 0; ph < n; ++ph) {
        int start = ph & 1;
        for (int i = tid; i < (n >> 1); i += NT) {
            int j = start + 2 * i;
            if (j + 1 < n && lam[j] < lam[j + 1]) {
                float tl = lam[j]; lam[j] = lam[j + 1]; lam[j + 1] = tl;
                int   to = ord[j]; ord[j] = ord[j + 1]; ord[j + 1] = to;
            }
        }
        __syncthreads();
    }
}

// ---------------------------------------------------------------------------
// MPS compression: forward orthogonalization sweep + backward truncation to CHI
// arena: 8 tensors T_y (bond[y], bond[y+1], 8) at y*SLOT.
// ---------------------------------------------------------------------------
__device__ void compress_mps(float* arena, int* bond,
                             float* Mb, float* Ub, float* Gb, float* EVb,
                             float* Wb, float* SB,
                             float* lam, int* ord,
                             float* shc, float* shs, int* shp, int* shq,
                             float* Ssrt)
{
    // ---- forward sweep (left-orthogonalize, no truncation) ----
    for (int i = 0; i < 7; ++i) {
        const int L = bond[i], R = bond[i + 1];
        const int m = L * 8, n = R;
        float* T = arena + i * SLOT;
        __syncthreads();
        // M[(l*8+u)*n + r] = T[(l*R + r)*8 + u]
        for (int id = threadIdx.x; id < m * n; id += NT) {
            int row = id / n, r = id - row * n;
            int l = row >> 3, u = row & 7;
            Mb[id] = T[(l * R + r) * 8 + u];
        }
        __syncthreads();
        int k;
        if (m >= n) {
            block_gemm(Gb, Mb, Mb, n, n, m, 1, n, n, 1, n, 1);   // G = M^T M
            __syncthreads();
            block_jacobi(Gb, EVb, n, shc, shs, shp, shq);
            __syncthreads();
            for (int j = threadIdx.x; j < n; j += NT)
                lam[j] = sqrtf(fmaxf(Gb[j * n + j], 0.f));
            __syncthreads();
            block_gemm(Ub, Mb, EVb, m, n, n, n, 1, n, 1, n, 1);  // U = M V
            __syncthreads();
            for (int id = threadIdx.x; id < m * n; id += NT) {
                int j = id % n;
                Ub[id] = Ub[id] / fmaxf(lam[j], 1e-20f);         // U S^-1
            }
            __syncthreads();
            // ts[i] <- U reshaped (L, n, 8)
            for (int id = threadIdx.x; id < m * n; id += NT) {
                int row = id / n, j = id - row * n;
                int l = row >> 3, u = row & 7;
                T[(l * n + j) * 8 + u] = Ub[id];
            }
            // W = diag(S) V^T
            for (int id = threadIdx.x; id < n * n; id += NT) {
                int j = id / n, c = id - j * n;
                Wb[id] = lam[j] * EVb[c * n + j];
            }
            k = n;
        } else {
            block_gemm(Gb, Mb, Mb, m, m, n, n, 1, 1, n, m, 1);   // G = M M^T
            __syncthreads();
            block_jacobi(Gb, EVb, m, shc, shs, shp, shq);
            __syncthreads();
            // ts[i] <- U (=EV) reshaped (L, m, 8)
            for (int id = threadIdx.x; id < m * m; id += NT) {
                int row = id / m, j = id - row * m;
                int l = row >> 3, u = row & 7;
                T[(l * m + j) * 8 + u] = EVb[id];
            }
            __syncthreads();
            block_gemm(Wb, EVb, Mb, m, n, m, 1, m, n, 1, n, 1);  // W = U^T M = S Vh
            k = m;
        }
        __syncthreads();
        // ts[i+1] <- W @ ts[i+1]
        const int L1 = R, R1 = bond[i + 2];
        float* T1 = arena + (i + 1) * SLOT;
        block_gemm(Mb, Wb, T1, k, R1 * 8, L1, L1, 1, R1 * 8, 1, R1 * 8, 1);
        __syncthreads();
        block_copy(T1, Mb, k * R1 * 8);
        __syncthreads();
        bond[i + 1] = k;
    }
    // ---- backward sweep (truncate to CHI) ----
    for (int i = 7; i >= 1; --i) {
        const int L = bond[i], Rr = bond[i + 1];
        const int m = L, n = Rr * 8;
        float* T = arena + i * SLOT;              // M = T directly (m x n)
        const int mn = m < n ? m : n;
        const int keep = mn < CHI ? mn : CHI;
        __syncthreads();
        if (m <= n) {
            block_gemm(Gb, T, T, m, m, n, n, 1, 1, n, m, 1);     // G = M M^T
            __syncthreads();
            block_jacobi(Gb, EVb, m, shc, shs, shp, shq);
            __syncthreads();
            sort_eigs(Gb, m, lam, ord);
            __syncthreads();
            for (int j = threadIdx.x; j < keep; j += NT)
                Ssrt[j] = sqrtf(fmaxf(lam[j], 0.f));
            __syncthreads();
            // Us (m x keep): sorted top eigenvectors
            for (int id = threadIdx.x; id < m * keep; id += NT) {
                int a = id / keep, j = id - a * keep;
                SB[id] = EVb[a * m + ord[j]];
            }
            __syncthreads();
            block_gemm(Ub, SB, T, keep, n, m, 1, keep, n, 1, n, 1); // Us^T M
            __syncthreads();
            for (int id = threadIdx.x; id < keep * n; id += NT) {
                int j = id / n;
                Ub[id] = Ub[id] / fmaxf(Ssrt[j], 1e-20f);        // Vh = S^-1 Us^T M
            }
            // Ck = Us * S
            for (int id = threadIdx.x; id < m * keep; id += NT) {
                int j = id % keep;
                Wb[id] = SB[id] * Ssrt[j];
            }
        } else {
            block_gemm(Gb, T, T, n, n, m, 1, n, n, 1, n, 1);     // G = M^T M
            __syncthreads();
            block_jacobi(Gb, EVb, n, shc, shs, shp, shq);
            __syncthreads();
            sort_eigs(Gb, n, lam, ord);
            __syncthreads();
            // Vs (n x keep)
            for (int id = threadIdx.x; id < n * keep; id += NT) {
                int c = id / keep, j = id - c * keep;
                SB[id] = EVb[c * n + ord[j]];
            }
            __syncthreads();
            // Vh = Vs^T
            for (int id = threadIdx.x; id < keep * n; id += NT) {
                int j = id / n, c = id - j * n;
                Ub[id] = SB[c * keep + j];
            }
            __syncthreads();
            block_gemm(Wb, T, SB, m, keep, n, n, 1, keep, 1, keep, 1); // Ck = M Vs
        }
        __syncthreads();
        // ts[i-1][a][k][u] = sum_mm old[a][mm][u] * Ck[mm][k]  (batched over u)
        const int L0 = bond[i - 1];
        float* T0 = arena + (i - 1) * SLOT;
        for (int u = 0; u < 8; ++u) {
            block_gemm(Mb + u, T0 + u, Wb, L0, keep, m,
                       m * 8, 8, keep, 1, keep * 8, 8);
        }
        __syncthreads();
        block_copy(T, Ub, keep * n);             // commit ts[i] = Vh (keep,Rr,8)
        __syncthreads();
        block_copy(T0, Mb, L0 * keep * 8);
        __syncthreads();
        bond[i] = keep;
    }
}

// ---------------------------------------------------------------------------
// Absorb one PEPS row into a boundary MPS (bottom: contract site d-index with
// MPS third index; top: contract site u-index).
// ---------------------------------------------------------------------------
__device__ void absorb_row(float* arena, int* bond, int xi, bool isBottom,
                           const int* xin, int sample, const float* peps,
                           float* siteBuf, float* Mb, float* Ub)
{
    // materialize sampled site tensors, compact (8,8,Ld,Rd)
    for (int y = 0; y < LYDIM; ++y) {
        const int s = xin[sample * (LXDIM * LYDIM) + xi * LYDIM + y];
        const int Ldy = (y == 0) ? 1 : 8, Rdy = (y == LYDIM - 1) ? 1 : 8;
        const int cnt = 8 * 8 * Ldy * Rdy;
        float* st = siteBuf + y * 4096;
        for (int id = threadIdx.x; id < cnt; id += NT) {
            int r = id % Rdy; int t1 = id / Rdy;
            int l = t1 % Ldy; int t2 = t1 / Ldy;
            int d = t2 & 7;   int u = t2 >> 3;
            size_t pidx = ((((((size_t)xi * LYDIM + y) * DDIM + u) * DDIM + d)
                            * DDIM + l) * DDIM + r) * PDIM + s;
            st[id] = peps[pidx];
        }
    }
    __syncthreads();
    for (int y = 0; y < LYDIM; ++y) {
        const int X = bond[y], Y = bond[y + 1];
        const int Ldy = (y == 0) ? 1 : 8, Rdy = (y == LYDIM - 1) ? 1 : 8;
        const int LR = Ldy * Rdy;
        const int YR = Y * Rdy;
        float* T = arena + y * SLOT;
        const float* st = siteBuf + y * 4096;
        if (isBottom) {
            for (int u = 0; u < 8; ++u) {
                // C_u[(x,y2)][(l,r)] = sum_d T[(x,y2),d] * site[u][d][(l,r)]
                block_gemm(Ub, T, st + u * 8 * LR, X * Y, LR, 8,
                           8, 1, LR, 1, LR, 1);
                __syncthreads();
                for (int id = threadIdx.x; id < X * Y * LR; id += NT) {
                    int xy = id / LR, j = id - xy * LR;
                    int xx = xy / Y, yy = xy - xx * Y;
                    int l = j / Rdy, r = j - l * Rdy;
                    Mb[((xx * Ldy + l) * YR + (yy * Rdy + r)) * 8 + u] = Ub[id];
                }
                __syncthreads();
            }
        } else {
            // C[(x,y2)][(d,l,r)] = sum_u T[(x,y2),u] * site[u][(d,l,r)]
            block_gemm(Ub, T, st, X * Y, 8 * LR, 8, 8, 1, 8 * LR, 1, 8 * LR, 1);
            __syncthreads();
            for (int id = threadIdx.x; id < X * Y * 8 * LR; id += NT) {
                int xy = id / (8 * LR), j = id - xy * (8 * LR);
                int xx = xy / Y, yy = xy - xx * Y;
                int d = j / LR, j2 = j - d * LR;
                int l = j2 / Rdy, r = j2 - l * Rdy;
                Mb[((xx * Ldy + l) * YR + (yy * Rdy + r)) * 8 + d] = Ub[id];
            }
            __syncthreads();
        }
        block_copy(T, Mb, X * Ldy * YR * 8);
        __syncthreads();
    }
    int nb[9];
    for (int y = 0; y < 8; ++y) nb[y] = bond[y] * ((y == 0) ? 1 : 8);
    nb[8] = bond[8];
    for (int y = 0; y < 9; ++y) bond[y] = nb[y];
}

// ---------------------------------------------------------------------------
// Main kernel: one block per sample.
// ---------------------------------------------------------------------------
__global__ __launch_bounds__(NT)
void peps_amp_kernel(const int* __restrict__ xin,
                     const float* __restrict__ peps,
                     float* __restrict__ out,
                     float* __restrict__ ws)
{
    const int sample = blockIdx.x;
    const int tid = threadIdx.x;
    float* base = ws + (size_t)sample * SAMPLE_F;
    float* botA = base + OFF_BOT;
    float* topA = base + OFF_TOP;
    float* Mb   = base + OFF_M;
    float* Ub   = base + OFF_U;
    float* Gb   = base + OFF_G;
    float* EVb  = base + OFF_EV;
    float* Wb   = base + OFF_W;
    float* SB   = base + OFF_SB;
    float* siteBuf = base + OFF_SITE;

    __shared__ float shc[128], shs[128];
    __shared__ int   shp[128], shq[128];
    __shared__ float lam[256];
    __shared__ int   ord[256];
    __shared__ float Ssrt[CHI];
    __shared__ float resA[CHI * CHI];
    __shared__ float resB[CHI * CHI];
    __shared__ float tmpC[CHI * CHI * 8];

    int bb[9], tb[9];
    for (int y = 0; y < 9; ++y) { bb[y] = 8; tb[y] = 8; }
    bb[0] = 1; bb[8] = 1; tb[0] = 1; tb[8] = 1;

    // ---- init bottom MPS from row 0: bot[y][l][r][u] = peps[0,y,u,0,l,r,s]
    for (int y = 0; y < LYDIM; ++y) {
        const int s = xin[sample * 64 + 0 * LYDIM + y];
        const int Ldy = (y == 0) ? 1 : 8, Rdy = (y == LYDIM - 1) ? 1 : 8;
        float* T = botA + y * SLOT;
        const int cnt = Ldy * Rdy * 8;
        for (int id = tid; id < cnt; id += NT) {
            int u = id & 7; int t1 = id >> 3;
            int r = t1 % Rdy; int l = t1 / Rdy;
            size_t pidx = ((((((size_t)0 * LYDIM + y) * DDIM + u) * DDIM + 0)
                            * DDIM + l) * DDIM + r) * PDIM + s;
            T[id] = peps[pidx];
        }
    }
    __syncthreads();
    for (int xi = 1; xi <= LXDIM / 2 - 1; ++xi) {
        absorb_row(botA, bb, xi, true, xin, sample, peps, siteBuf, Mb, Ub);
        __syncthreads();
        compress_mps(botA, bb, Mb, Ub, Gb, EVb, Wb, SB,
                     lam, ord, shc, shs, shp, shq, Ssrt);
        __syncthreads();
    }

    // ---- init top MPS from row Lx-1: top[y][l][r][d] = peps[7,y,0,d,l,r,s]
    for (int y = 0; y < LYDIM; ++y) {
        const int s = xin[sample * 64 + (LXDIM - 1) * LYDIM + y];
        const int Ldy = (y == 0) ? 1 : 8, Rdy = (y == LYDIM - 1) ? 1 : 8;
        float* T = topA + y * SLOT;
        const int cnt = Ldy * Rdy * 8;
        for (int id = tid; id < cnt; id += NT) {
            int d = id & 7; int t1 = id >> 3;
            int r = t1 % Rdy; int l = t1 / Rdy;
            size_t pidx = ((((((size_t)(LXDIM - 1) * LYDIM + y) * DDIM + 0)
                            * DDIM + d) * DDIM + l) * DDIM + r) * PDIM + s;
            T[id] = peps[pidx];
        }
    }
    __syncthreads();
    for (int xi = LXDIM - 2; xi >= LXDIM / 2; --xi) {
        absorb_row(topA, tb, xi, false, xin, sample, peps, siteBuf, Mb, Ub);
        __syncthreads();
        compress_mps(topA, tb, Mb, Ub, Gb, EVb, Wb, SB,
                     lam, ord, shc, shs, shp, shq, Ssrt);
        __syncthreads();
    }

    // ---- final zipper contraction over y ----
    int a = bb[1], c = tb[1];
    {
        const float* B0 = botA + 0 * SLOT;   // (1, a, 8)
        const float* T0 = topA + 0 * SLOT;   // (1, c, 8)
        for (int id = tid; id < a * c; id += NT) {
            int b = id / c, d = id - b * c;
            float acc = 0.f;
            for (int kk = 0; kk < 8; ++kk)
                acc += B0[b * 8 + kk] * T0[d * 8 + kk];
            resA[id] = acc;
        }
    }
    __syncthreads();
    for (int y = 1; y < LYDIM; ++y) {
        const int b2 = bb[y + 1], d2 = tb[y + 1];
        const float* By = botA + y * SLOT;   // (a, b2, 8)
        const float* Ty = topA + y * SLOT;   // (c, d2, 8)
        // tmp[cc][bb'][k] = sum_a res[a][cc] * By[a][bb'][k]
        for (int id = tid; id < c * b2 * 8; id += NT) {
            int cc = id / (b2 * 8), rem = id - cc * (b2 * 8);
            int bbq = rem >> 3, kk = rem & 7;
            float acc = 0.f;
            for (int aa = 0; aa < a; ++aa)
                acc += resA[aa * c + cc] * By[(aa * b2 + bbq) * 8 + kk];
            tmpC[id] = acc;
        }
        __syncthreads();
        // res'[bb'][dd'] = sum_{cc,k} tmp[cc][bb'][k] * Ty[cc][dd'][k]
        for (int id = tid; id < b2 * d2; id += NT) {
            int bbq = id / d2, dd = id - bbq * d2;
            float acc = 0.f;
            for (int cc = 0; cc < c; ++cc)
                for (int kk = 0; kk < 8; ++kk)
                    acc += tmpC[cc * b2 * 8 + bbq * 8 + kk]
                         * Ty[(cc * d2 + dd) * 8 + kk];
            resB[id] = acc;
        }
        __syncthreads();
        for (int id = tid; id < b2 * d2; id += NT) resA[id] = resB[id];
        __syncthreads();
        a = b2; c = d2;
    }
    if (tid == 0) out[sample] = resA[0];
}

extern "C" void kernel_launch(void* const* d_in, const int* in_sizes, int n_in,
                              void* d_out, int out_size, void* d_ws, size_t ws_size,
                              hipStream_t stream)
{
    const int*   xin  = (const int*)d_in[0];
    const float* peps = (const float*)d_in[1];
    float* out = (float*)d_out;
    float* ws  = (float*)d_ws;
    const int nsamp = in_sizes[0] / (LXDIM * LYDIM);
    if (nsamp <= 0) return;
    hipLaunchKernelGGL(peps_amp_kernel, dim3(nsamp), dim3(NT), 0, stream,
                       xin, peps, out, ws);
}